// SelfAttention3d_12025908429013
// MI455X (gfx1250) — compile-verified
//
#include <hip/hip_runtime.h>
#include <hip/hip_bf16.h>

typedef __attribute__((ext_vector_type(2))) float v2f;
typedef __attribute__((ext_vector_type(8))) float v8f;

#define NTOK 4096
#define CCH  64
#define CKCH 8
#define NJT  (NTOK / 16)

// ---------------------------------------------------------------------------
// Tensor Data Mover support (gfx1250).  Guarded so a toolchain without the
// builtin falls back to cooperative VMEM->LDS loads.
// ---------------------------------------------------------------------------
#if defined(__gfx1250__) && __has_builtin(__builtin_amdgcn_tensor_load_to_lds)
#define HAVE_TDM 1
#else
#define HAVE_TDM 0
#endif

#if HAVE_TDM
typedef unsigned int u32x4 __attribute__((ext_vector_type(4)));
typedef int          i32x4 __attribute__((ext_vector_type(4)));
typedef int          i32x8 __attribute__((ext_vector_type(8)));

// Issue one 2D tile DMA: global (row-major, row_stride elems of 4B) -> LDS,
// rows packed contiguously (tile_w floats per row).
__device__ __forceinline__ void tdm_load_tile_2d(
    unsigned lds_byte_addr, const float* gptr,
    unsigned tile_w, unsigned tile_h,
    unsigned tensor_w, unsigned tensor_h, unsigned row_stride_elems)
{
    unsigned long long ga = (unsigned long long)(size_t)gptr;
    u32x4 g0;
    g0[0] = 1u;                                        // count=1 (valid user D#)
    g0[1] = lds_byte_addr;                             // lds_addr
    g0[2] = (unsigned)(ga & 0xFFFFFFFFu);              // global_addr[31:0]
    g0[3] = (unsigned)((ga >> 32) & 0x1FFFFFFu)        // global_addr[56:32]
          | (2u << 30);                                // type = 2 ("image")
    i32x8 g1;
    g1[0] = (int)(2u << 16);                           // data_size=2 (4B), mask=0
    g1[1] = (int)((tensor_w & 0xFFFFu) << 16);         // tensor_dim0[15:0]
    g1[2] = (int)((tensor_w >> 16) |
                  ((tensor_h & 0xFFFFu) << 16));       // dim0[31:16] | dim1[15:0]
    g1[3] = (int)((tensor_h >> 16) |
                  ((tile_w & 0xFFFFu) << 16));         // dim1[31:16] | tile_dim0
    g1[4] = (int)(tile_h & 0xFFFFu);                   // tile_dim1 (tile_dim2=0)
    g1[5] = (int)row_stride_elems;                     // tensor_dim0_stride[31:0]
    g1[6] = 0;                                         // stride0[47:32]|stride1 lo
    g1[7] = 0;
    i32x4 z4 = {0, 0, 0, 0};
#if __clang_major__ >= 23
    i32x8 z8 = {0, 0, 0, 0, 0, 0, 0, 0};
    __builtin_amdgcn_tensor_load_to_lds(g0, g1, z4, z4, z8, 0);
#else
    __builtin_amdgcn_tensor_load_to_lds(g0, g1, z4, z4, 0);
#endif
}
#endif // HAVE_TDM

// ---------------------------------------------------------------------------
// Kernel 1: 1x1x1 conv projections q/k/v.  One thread per (b, n) column.
// Weight reads are lane-uniform -> scalar loads, broadcast from L2.
// ---------------------------------------------------------------------------
__global__ void __launch_bounds__(256) qkv_proj_kernel(
    const float* __restrict__ x,
    const float* __restrict__ wq, const float* __restrict__ bq,
    const float* __restrict__ wk, const float* __restrict__ bk,
    const float* __restrict__ wv, const float* __restrict__ bv,
    float* __restrict__ qw, float* __restrict__ kw, float* __restrict__ vw)
{
    int g = blockIdx.x * 256 + threadIdx.x;   // 0 .. B*N-1
    int b = g >> 12;
    int n = g & (NTOK - 1);

    float xc[CCH];
    const float* xb = x + (size_t)(b * CCH) * NTOK + n;
#pragma unroll
    for (int c = 0; c < CCH; ++c) xc[c] = xb[(size_t)c * NTOK];

#pragma unroll
    for (int o = 0; o < CKCH; ++o) {
        float aq = bq[o];
        float ak = bk[o];
#pragma unroll
        for (int c = 0; c < CCH; ++c) {
            aq = fmaf(wq[o * CCH + c], xc[c], aq);
            ak = fmaf(wk[o * CCH + c], xc[c], ak);
        }
        qw[(size_t)(b * CKCH + o) * NTOK + n] = aq;
        kw[(size_t)(b * CKCH + o) * NTOK + n] = ak;
    }

#pragma unroll 4
    for (int o = 0; o < CCH; ++o) {
        float av = bv[o];
#pragma unroll
        for (int c = 0; c < CCH; ++c) av = fmaf(wv[o * CCH + c], xc[c], av);
        vw[(size_t)(b * CCH + o) * NTOK + n] = av;
    }
}

// ---------------------------------------------------------------------------
// Kernel 2: flash attention with V_WMMA_F32_16X16X4_F32 + TDM tile streaming.
// Block = 128 threads = 4 waves; each wave owns a 16-query tile; the block
// shares double-buffered K/V tiles in LDS.  Wave 0 (selected via a
// readfirstlane-uniform predicate so the compiler emits a scalar branch, not
// EXEC masking -- TDM ops ignore EXEC) drives the Tensor Data Mover: while
// all waves run WMMAs on buffer p, the DMA for tile t+1 fills buffer p^1
// (tracked with TENSORcnt).
// ---------------------------------------------------------------------------
__global__ void __launch_bounds__(128) flash_attn_kernel(
    const float* __restrict__ x,
    const float* __restrict__ qw, const float* __restrict__ kw,
    const float* __restrict__ vw, const float* __restrict__ gamma,
    float* __restrict__ out)
{
    __shared__ float ktile[2][CKCH * 16];   // K tile  [ck][jj]
    __shared__ float vtile[2][CCH * 16];    // V tile  [c][jj]
    __shared__ float pt[4][16 * 16];        // per-wave P tile [j][i]

    const int tid  = threadIdx.x;           // 0..127
    const int lane = tid & 31;
    const int wave = tid >> 5;              // 0..3
    const int b    = blockIdx.y;
    const int it0  = blockIdx.x * 64 + wave * 16;   // query-tile base

    const int iloc  = lane & 15;            // query index within tile (N of D)
    const int kb    = (lane >> 4) << 1;     // K-operand base: 0 / 2
    const int half8 = (lane >> 4) << 3;     // M offset: 0 / 8

    // Wave-uniform predicate in an SGPR -> scalar branch around TDM blocks.
    const bool tdmWave = (__builtin_amdgcn_readfirstlane(wave) == 0);

    // Q as fixed B-matrix operand (ck x i), split into two K=4 groups.
    const float* qbp = qw + (size_t)(b * CKCH) * NTOK + it0 + iloc;
    v2f qB0, qB1;
    qB0.x = qbp[(size_t)(kb + 0) * NTOK];
    qB0.y = qbp[(size_t)(kb + 1) * NTOK];
    qB1.x = qbp[(size_t)(kb + 4) * NTOK];
    qB1.y = qbp[(size_t)(kb + 5) * NTOK];

    v8f acc0 = {}, acc1 = {}, acc2 = {}, acc3 = {};
    float m = -3.0e38f;
    float l = 0.0f;

    const float* kwb = kw + (size_t)(b * CKCH) * NTOK;
    const float* vwb = vw + (size_t)(b * CCH) * NTOK;
    float* pw = &pt[wave][0];

#if HAVE_TDM
    if (tdmWave) {   // prologue: DMA tile 0 into buffer 0
        tdm_load_tile_2d((unsigned)(size_t)&ktile[0][0], kwb,
                         16, CKCH, NTOK, CKCH, NTOK);
        tdm_load_tile_2d((unsigned)(size_t)&vtile[0][0], vwb,
                         16, CCH, NTOK, CCH, NTOK);
    }
#endif

    for (int t = 0; t < NJT; ++t) {
        const int p  = t & 1;
        const int j0 = t * 16;
        const float* kT = &ktile[p][0];
        const float* vT = &vtile[p][0];

#if HAVE_TDM
        if (tdmWave) __builtin_amdgcn_s_wait_tensorcnt((short)0);
        __syncthreads();                       // tile t visible to all waves
        if (tdmWave && (t + 1 < NJT)) {        // DMA tile t+1 behind compute
            const int jn = j0 + 16;
            tdm_load_tile_2d((unsigned)(size_t)&ktile[p ^ 1][0], kwb + jn,
                             16, CKCH, NTOK, CKCH, NTOK);
            tdm_load_tile_2d((unsigned)(size_t)&vtile[p ^ 1][0], vwb + jn,
                             16, CCH, NTOK, CCH, NTOK);
        }
#else
        __syncthreads();                       // previous tile fully consumed
        {
            int ck = tid >> 4, jj = tid & 15;
            ktile[p][ck * 16 + jj] = kwb[(size_t)ck * NTOK + j0 + jj];
        }
#pragma unroll
        for (int u = 0; u < 8; ++u) {
            int idx = tid + 128 * u;
            int c = idx >> 4, jj = idx & 15;
            vtile[p][c * 16 + jj] = vwb[(size_t)c * NTOK + j0 + jj];
        }
        __syncthreads();
        if (j0 + 16 < NTOK)
            __builtin_prefetch(vwb + (size_t)(tid >> 4) * NTOK + j0 + 16, 0, 0);
#endif

        // ---- energy tile, transposed: E[j,i] = sum_ck K[j,ck] * Q[ck,i] ----
        v2f kA0, kA1;
        kA0.x = kT[(kb + 0) * 16 + iloc];
        kA0.y = kT[(kb + 1) * 16 + iloc];
        kA1.x = kT[(kb + 4) * 16 + iloc];
        kA1.y = kT[(kb + 5) * 16 + iloc];
        v8f e = {};
        e = __builtin_amdgcn_wmma_f32_16x16x4_f32(false, kA0, false, qB0,
                                                  (short)0, e, false, false);
        e = __builtin_amdgcn_wmma_f32_16x16x4_f32(false, kA1, false, qB1,
                                                  (short)0, e, false, false);

        // ---- online softmax over j (VGPR rows + cross-half shuffle) ----
        float tm = e[0];
#pragma unroll
        for (int r = 1; r < 8; ++r) tm = fmaxf(tm, e[r]);
        tm = fmaxf(tm, __shfl_xor(tm, 16, 32));
        float mn = fmaxf(m, tm);
        float sc = __expf(m - mn);
        m = mn;

        float pr[8];
        float ps = 0.0f;
#pragma unroll
        for (int r = 0; r < 8; ++r) {
            pr[r] = __expf(e[r] - mn);
            ps += pr[r];
        }
        ps += __shfl_xor(ps, 16, 32);
        l = l * sc + ps;

#pragma unroll
        for (int r = 0; r < 8; ++r) {
            acc0[r] *= sc; acc1[r] *= sc; acc2[r] *= sc; acc3[r] *= sc;
        }

        // ---- stage P through wave-private LDS to build the B operand ----
#pragma unroll
        for (int r = 0; r < 8; ++r)
            pw[(half8 + r) * 16 + iloc] = pr[r];   // P[j][i]

        // ---- PV: O[c,i] += V[c,j] * P[j,i], K=4 per step, 4 c-tiles ----
#pragma unroll
        for (int s = 0; s < 4; ++s) {
            const int jb = 4 * s + kb;
            v2f pB;
            pB.x = pw[(jb + 0) * 16 + iloc];
            pB.y = pw[(jb + 1) * 16 + iloc];
            v2f vA0, vA1, vA2, vA3;
            vA0.x = vT[( 0 + iloc) * 16 + jb]; vA0.y = vT[( 0 + iloc) * 16 + jb + 1];
            vA1.x = vT[(16 + iloc) * 16 + jb]; vA1.y = vT[(16 + iloc) * 16 + jb + 1];
            vA2.x = vT[(32 + iloc) * 16 + jb]; vA2.y = vT[(32 + iloc) * 16 + jb + 1];
            vA3.x = vT[(48 + iloc) * 16 + jb]; vA3.y = vT[(48 + iloc) * 16 + jb + 1];
            acc0 = __builtin_amdgcn_wmma_f32_16x16x4_f32(false, vA0, false, pB,
                                                         (short)0, acc0, false, false);
            acc1 = __builtin_amdgcn_wmma_f32_16x16x4_f32(false, vA1, false, pB,
                                                         (short)0, acc1, false, false);
            acc2 = __builtin_amdgcn_wmma_f32_16x16x4_f32(false, vA2, false, pB,
                                                         (short)0, acc2, false, false);
            acc3 = __builtin_amdgcn_wmma_f32_16x16x4_f32(false, vA3, false, pB,
                                                         (short)0, acc3, false, false);
        }
    }

    // ---- epilogue: normalize, residual, gamma ----
    const float invl = 1.0f / l;
    const float gm   = gamma[0];
    const float* xb  = x   + (size_t)(b * CCH) * NTOK;
    float*       ob  = out + (size_t)(b * CCH) * NTOK;
    const int n = it0 + iloc;
#pragma unroll
    for (int r = 0; r < 8; ++r) {
        int c;
        c =  0 + half8 + r;
        ob[(size_t)c * NTOK + n] = gm * (acc0[r] * invl) + xb[(size_t)c * NTOK + n];
        c = 16 + half8 + r;
        ob[(size_t)c * NTOK + n] = gm * (acc1[r] * invl) + xb[(size_t)c * NTOK + n];
        c = 32 + half8 + r;
        ob[(size_t)c * NTOK + n] = gm * (acc2[r] * invl) + xb[(size_t)c * NTOK + n];
        c = 48 + half8 + r;
        ob[(size_t)c * NTOK + n] = gm * (acc3[r] * invl) + xb[(size_t)c * NTOK + n];
    }
}

// ---------------------------------------------------------------------------
extern "C" void kernel_launch(void* const* d_in, const int* in_sizes, int n_in,
                              void* d_out, int out_size, void* d_ws, size_t ws_size,
                              hipStream_t stream) {
    const float* x     = (const float*)d_in[0];
    const float* wq    = (const float*)d_in[1];
    const float* bq    = (const float*)d_in[2];
    const float* wk    = (const float*)d_in[3];
    const float* bk    = (const float*)d_in[4];
    const float* wv    = (const float*)d_in[5];
    const float* bv    = (const float*)d_in[6];
    const float* gamma = (const float*)d_in[7];
    float* out = (float*)d_out;

    // Workspace layout: q [4*8*4096] | k [4*8*4096] | v [4*64*4096] floats.
    float* qw = (float*)d_ws;
    float* kw = qw + 4 * CKCH * NTOK;
    float* vw = kw + 4 * CKCH * NTOK;

    dim3 g1((4 * NTOK) / 256);
    qkv_proj_kernel<<<g1, 256, 0, stream>>>(x, wq, bq, wk, bk, wv, bv, qw, kw, vw);

    dim3 g2(NTOK / 64, 4);
    flash_attn_kernel<<<g2, 128, 0, stream>>>(x, qw, kw, vw, gamma, out);
}